// ARModel_13305808683645
// MI455X (gfx1250) — compile-verified
//
#include <hip/hip_runtime.h>
#include <math.h>

// AR(16) log-likelihood:  logp[b][t] = -0.5*log(2*pi*sd^2) - 0.5*((s_t - pred_t)/sd)^2
//   pred_t = sum_{k=1..16} coeffs[k-1] * s[t-k]   (zeros before t=0)
//
// Strategy (MI455X / gfx1250):
//   * 16x16 output tiles (16 batch rows x 16 timesteps) computed as a banded GEMM
//     D = A(16x32 window) x B(32x16 Toeplitz of coeffs) with 8 chained
//     V_WMMA_F32_16X16X4_F32 ops -> full f32 precision, dot products off the VALU.
//   * Each wave owns 16 rows x 256 timesteps; window kept in registers and
//     rotated so every s element is read from HBM once for the A path.
//   * Kernel is HBM-bound: ~64 MiB read + 64 MiB write  (~5.5 us floor @ 23.3 TB/s).

typedef __attribute__((ext_vector_type(2))) float v2f;
typedef __attribute__((ext_vector_type(8))) float v8f;

#define T_LEN   65536
#define B_ROWS  256
#define P_COEF  16
#define ROWG    (B_ROWS / 16)       // 16 row groups
#define CHUNKS  256                 // time chunks per row group
#define CHUNK_T (T_LEN / CHUNKS)    // 256 timesteps = 16 tiles per wave

__global__ __launch_bounds__(32) void ar16_logp_wmma(
    const float* __restrict__ s,       // [B_ROWS][T_LEN]
    const float* __restrict__ coeffs,  // [P_COEF]
    const float* __restrict__ noise_std,
    float* __restrict__ out)           // [B_ROWS][T_LEN]
{
    const int lane = threadIdx.x;      // 0..31 (wave32)
    const int n    = lane & 15;        // N for B/C/D, M for A
    const int h    = lane >> 4;        // half-wave select (K pair / M+8)

    const int rg    = blockIdx.x & (ROWG - 1);
    const int chunk = blockIdx.x >> 4;            // needs ROWG == 16
    const int t0c   = chunk * CHUNK_T;

    const float sd      = noise_std[0];
    const float inv_var = 1.0f / (sd * sd);
    const float c0      = -0.5f * logf(6.283185307179586f * sd * sd);

    // ---- B: constant 32x16 banded Toeplitz of coeffs, in WMMA B layout ----
    // chunk c covers K rows j = 4c..4c+3; this lane holds j0 = 4c+2h and j0+1.
    // B[j][n] = coeffs[15+n-j] if 0 <= 15+n-j < 16 else 0.
    v2f b[8];
#pragma unroll
    for (int c = 0; c < 8; ++c) {
        int j0 = 4 * c + 2 * h;
        int i0 = 15 + n - j0;
        int i1 = i0 - 1;
        float v0 = (i0 >= 0 && i0 < P_COEF) ? coeffs[i0] : 0.0f;
        float v1 = (i1 >= 0 && i1 < P_COEF) ? coeffs[i1] : 0.0f;
        b[c] = (v2f){v0, v1};
    }

    // ---- A window: 8 chunks of float2 per lane; lane's A-row is m = n ----
    // chunk c holds s[row][t0-16 + 4c + 2h (+1)]
    const float* rowA = s + (size_t)(rg * 16 + n) * T_LEN;
    v2f a[8];
    if (t0c == 0) {
#pragma unroll
        for (int c = 0; c < 4; ++c) a[c] = (v2f){0.0f, 0.0f};   // left zero-pad
    } else {
#pragma unroll
        for (int c = 0; c < 4; ++c)
            a[c] = *(const v2f*)(rowA + (t0c - 16) + 4 * c + 2 * h);
    }

    const int erow = rg * 16 + 8 * h;  // epilogue rows erow+r, r = 0..7

    for (int t0 = t0c; t0 < t0c + CHUNK_T; t0 += 16) {
        // New half of the window = s[row][t0 .. t0+15]
#pragma unroll
        for (int c = 0; c < 4; ++c)
            a[4 + c] = *(const v2f*)(rowA + t0 + 4 * c + 2 * h);

        // D = A x B  (8 x K=4 steps, f32 accumulate)
        v8f acc = {};
#pragma unroll
        for (int c = 0; c < 8; ++c)
            acc = __builtin_amdgcn_wmma_f32_16x16x4_f32(
                false, a[c], false, b[c], (short)0, acc, false, false);

        // Epilogue: lane holds D[m=erow+r][t0+n] in acc[r]; s re-read hits L0.
#pragma unroll
        for (int r = 0; r < 8; ++r) {
            size_t o  = (size_t)(erow + r) * T_LEN + (size_t)(t0 + n);
            float sv  = s[o];
            float d   = sv - acc[r];
            out[o]    = c0 - 0.5f * inv_var * d * d;
        }

        // Rotate window: old K=16..31 becomes new K=0..15
#pragma unroll
        for (int c = 0; c < 4; ++c) a[c] = a[4 + c];
    }
}

extern "C" void kernel_launch(void* const* d_in, const int* in_sizes, int n_in,
                              void* d_out, int out_size, void* d_ws, size_t ws_size,
                              hipStream_t stream) {
    (void)in_sizes; (void)n_in; (void)out_size; (void)d_ws; (void)ws_size;
    const float* s      = (const float*)d_in[0];
    const float* coeffs = (const float*)d_in[1];
    const float* nsd    = (const float*)d_in[2];
    float* out          = (float*)d_out;

    dim3 grid(ROWG * CHUNKS);   // 4096 single-wave blocks
    dim3 block(32);
    ar16_logp_wmma<<<grid, block, 0, stream>>>(s, coeffs, nsd, out);
}